// DividPart_65274912965111
// MI455X (gfx1250) — compile-verified
//
#include <hip/hip_runtime.h>
#include <stdint.h>

#define FRAMES 128
#define JOINTS 17
#define NPARTS 7
#define ROWF   (FRAMES * JOINTS)          // 2176 floats staged per batch
#define CHAN_STRIDE  (FRAMES * JOINTS)    // offset of channel 1 within a batch
#define BATCH_STRIDE (3 * FRAMES * JOINTS)
#define NWAVES (FRAMES / 32)

typedef unsigned int u32x4 __attribute__((ext_vector_type(4)));
typedef int          i32x4 __attribute__((ext_vector_type(4)));
typedef int          i32x8 __attribute__((ext_vector_type(8)));

// Issue one TDM DMA: 2176 contiguous f32 from global -> LDS.
// D# layout per CDNA5 ISA ch.8 (group0 128b, group1 256b; groups 2/3 zero for 2D).
__device__ __forceinline__ void tdm_load_rows(uint32_t lds_byte_off,
                                              const float* __restrict__ gsrc) {
  uint64_t ga = (uint64_t)(uintptr_t)(const void*)gsrc;
  u32x4 g0;
  g0.x = 1u;                                   // count=1 (valid), user mode
  g0.y = lds_byte_off;                         // lds_addr [63:32]
  g0.z = (uint32_t)ga;                         // global_addr lo32
  g0.w = (uint32_t)((ga >> 32) & 0x01FFFFFFu)  // global_addr [56:32]
       | (2u << 30);                           // type=2 ("image")
  i32x8 g1;
  g1[0] = (int)(2u << 16);                     // workgroup_mask=0, data_size=2 (4B)
  g1[1] = (int)((ROWF & 0xFFFFu) << 16);       // tensor_dim0 lo16 (=2176)
  g1[2] = (int)(((ROWF >> 16) & 0xFFFFu) | (1u << 16)); // dim0 hi16, tensor_dim1=1 lo16
  g1[3] = (int)((unsigned)ROWF << 16);         // tensor_dim1 hi16=0, tile_dim0=2176
  g1[4] = 1;                                   // tile_dim1=1, tile_dim2=0
  g1[5] = ROWF;                                // tensor_dim0_stride lo32
  g1[6] = (int)((ROWF & 0xFFFFu) << 16);       // stride0 hi16=0, stride1 lo16
  g1[7] = (int)(ROWF >> 16);                   // stride1 hi bits
  i32x4 z4 = {0, 0, 0, 0};
#if defined(__clang_major__) && (__clang_major__ >= 23)
  i32x8 z8 = {0, 0, 0, 0, 0, 0, 0, 0};
  __builtin_amdgcn_tensor_load_to_lds(g0, g1, z4, z4, z8, 0);
#else
  __builtin_amdgcn_tensor_load_to_lds(g0, g1, z4, z4, 0);
#endif
}

__global__ __launch_bounds__(FRAMES)
void DividPart_kernel(const float* __restrict__ poses, int* __restrict__ out, int B) {
  __shared__ __align__(16) float stage[ROWF];
  __shared__ float red_mx[NWAVES][NPARTS];
  __shared__ float red_mn[NWAVES][NPARTS];

  const int b    = blockIdx.x;
  const int tid  = threadIdx.x;
  const int lane = tid & 31;
  const int wave = tid >> 5;

  // Wave 0 DMAs the batch's channel-1 tile into LDS, waits TENSORcnt, then
  // the workgroup barrier publishes it to all 4 waves.
  if (tid < 32) {
    tdm_load_rows((uint32_t)(uintptr_t)(void*)stage,
                  poses + (size_t)b * BATCH_STRIDE + CHAN_STRIDE);
    __builtin_amdgcn_s_wait_tensorcnt(0);
  }
  __syncthreads();

  // One frame per thread. Stride-17 dword reads: gcd(17,64)=1 -> conflict-free.
  const float* row = &stage[tid * JOINTS];
  float p[JOINTS];
  const float j0 = row[0];
  const float ratio = 0.5f * ((row[5] - j0) + (row[6] - j0));
  float rmin = __builtin_inff();
#pragma unroll
  for (int j = 0; j < JOINTS; ++j) {
    float v = (row[j] - j0) / ratio;   // keep true division to match reference
    p[j] = v;
    rmin = fminf(rmin, v);
  }

  const int part_of[JOINTS] = {0,0,0,0,0, 1,1, 2,2, 3,3, 4,4, 5,5, 6,6};
  float pmx[NPARTS], pmn[NPARTS];
#pragma unroll
  for (int q = 0; q < NPARTS; ++q) { pmx[q] = -__builtin_inff(); pmn[q] = __builtin_inff(); }
#pragma unroll
  for (int j = 0; j < JOINTS; ++j) {
    float v = p[j] - rmin;
    int q = part_of[j];
    pmx[q] = fmaxf(pmx[q], v);
    pmn[q] = fminf(pmn[q], v);
  }

  // wave32 butterfly reduction of the 7 (max,min) pairs
#pragma unroll
  for (int off = 16; off > 0; off >>= 1) {
#pragma unroll
    for (int q = 0; q < NPARTS; ++q) {
      pmx[q] = fmaxf(pmx[q], __shfl_xor(pmx[q], off, 32));
      pmn[q] = fminf(pmn[q], __shfl_xor(pmn[q], off, 32));
    }
  }
  if (lane == 0) {
#pragma unroll
    for (int q = 0; q < NPARTS; ++q) {
      red_mx[wave][q] = pmx[q];
      red_mn[wave][q] = pmn[q];
    }
  }
  __syncthreads();

  if (tid == 0) {
    float mx[NPARTS], mn[NPARTS];
#pragma unroll
    for (int q = 0; q < NPARTS; ++q) {
      float a = red_mx[0][q], c = red_mn[0][q];
#pragma unroll
      for (int w = 1; w < NWAVES; ++w) {
        a = fmaxf(a, red_mx[w][q]);
        c = fminf(c, red_mn[w][q]);
      }
      mx[q] = a; mn[q] = c;
    }
    // parts tile joints 0..16, so global extrema = extrema of part extrema
    float bottom = -__builtin_inff(), top = __builtin_inff();
#pragma unroll
    for (int q = 0; q < NPARTS; ++q) {
      bottom = fmaxf(bottom, mx[q]);
      top    = fminf(top,    mn[q]);
    }
    const float denom = bottom - top;
#pragma unroll
    for (int q = 0; q < NPARTS; ++q) {
      int ma = (int)ceilf((mx[q] - top) / denom * 64.0f);
      int mi = (int)floorf((mn[q] - top) / denom * 64.0f);
      const int hi = (q + 1) * 9;
      const int lo = q * 9;
      if (ma <= mi)      { ma = hi; mi = lo; }
      if (ma - mi > 30)  { ma = hi; mi = lo; }
      out[q * B + b]              = ma;   // ma block: shape (7, B)
      out[NPARTS * B + q * B + b] = mi;   // mi block: shape (7, B)
    }
  }
}

extern "C" void kernel_launch(void* const* d_in, const int* in_sizes, int n_in,
                              void* d_out, int out_size, void* d_ws, size_t ws_size,
                              hipStream_t stream) {
  const float* poses = (const float*)d_in[0];
  int* out = (int*)d_out;                 // reference output dtype is int32
  const int B = in_sizes[0] / BATCH_STRIDE;   // 4096
  DividPart_kernel<<<dim3(B), dim3(FRAMES), 0, stream>>>(poses, out, B);
}